// GCNLayer_80633716015334
// MI455X (gfx1250) — compile-verified
//
#include <hip/hip_runtime.h>
#include <hip/hip_bf16.h>

#define N_NODES   40000
#define FEATS     128
#define TILE      16

typedef __attribute__((ext_vector_type(2))) float v2f;
typedef __attribute__((ext_vector_type(8))) float v8f;

// ---------------------------------------------------------------------------
// Kernel 1: out[n, o] = sum_k h[n,k] * W[o,k] + b[o]
// One wave -> one 16x16 output tile via V_WMMA_F32_16X16X4_F32.
// Block = 256 threads (8 waves) covers the full 128-wide N dimension of one
// 16-row M tile; grid.x = 40000/16 = 2500 M tiles.
//
// Fragment layouts (CDNA5 ISA 7.12.2, fp32):
//   A 16x4 (MxK): lane L holds M = L&15; VGPR0 = K{0|2}, VGPR1 = K{1|3}
//                 (K base = (L>>4)*2)  -> per-lane float2 at h[m][k0 + kb]
//   B 4x16 (KxN): lane L holds N = L&15 with the same K split
//                 B[k][n] = W[n][k]    -> per-lane float2 at W[n][k0 + kb]
//   C/D 16x16:    VGPR r: lanes 0-15 -> M=r, lanes 16-31 -> M=r+8, N = L&15
// ---------------------------------------------------------------------------
__global__ __launch_bounds__(256) void gcn_gemm_wmma(
    const float* __restrict__ h, const float* __restrict__ W,
    const float* __restrict__ bias, float* __restrict__ out) {
  const int lane = threadIdx.x & 31;
  const int wave = threadIdx.x >> 5;        // N tile index 0..7
  const int m0   = blockIdx.x * TILE;       // M tile base
  const int n0   = wave * TILE;             // N tile base

  const int l15  = lane & 15;
  const int half = lane >> 4;               // 0: K={0,1}, 1: K={2,3}
  const int kb   = half * 2;

  const float* aptr = h + (size_t)(m0 + l15) * FEATS + kb;   // A: rows of h
  const float* bptr = W + (size_t)(n0 + l15) * FEATS + kb;   // B: rows of W (= cols of W^T)

  v8f acc = {};
#pragma unroll
  for (int k0 = 0; k0 < FEATS; k0 += 4) {
    v2f a = *reinterpret_cast<const v2f*>(aptr + k0);   // 8B aligned (kb,k0 even)
    v2f b = *reinterpret_cast<const v2f*>(bptr + k0);
    // 8 args: (neg_a, A, neg_b, B, c_mod, C, reuse_a, reuse_b)
    acc = __builtin_amdgcn_wmma_f32_16x16x4_f32(
        false, a, false, b, (short)0, acc, false, false);
  }

  const float bn = bias[n0 + l15];
  float* optr = out + (size_t)(m0 + half * 8) * FEATS + n0 + l15;
#pragma unroll
  for (int r = 0; r < 8; ++r) {
    optr[(size_t)r * FEATS] = acc[r] + bn;
  }
}

// ---------------------------------------------------------------------------
// Kernel 2: zero agg (d_out is poisoned by the harness; we must re-zero each
// call so the scatter-add is deterministic). float4 stores, grid-stride.
// ---------------------------------------------------------------------------
__global__ __launch_bounds__(256) void zero_f32(float* __restrict__ p, long n4) {
  long i = (long)blockIdx.x * blockDim.x + threadIdx.x;
  const long stride = (long)gridDim.x * blockDim.x;
  float4* p4 = reinterpret_cast<float4*>(p);
  const float4 z = make_float4(0.f, 0.f, 0.f, 0.f);
  for (; i < n4; i += stride) p4[i] = z;
}

// ---------------------------------------------------------------------------
// Hardware fp32 atomic add, device scope, no return value (STOREcnt-tracked;
// s_endpgm's implicit wait-idle drains it). Avoids clang's CAS-loop expansion
// of atomicrmw fadd when fine-grained-memory safety can't be proven.
// ---------------------------------------------------------------------------
__device__ __forceinline__ void atomic_add_f32_dev(float* p, float v) {
  asm volatile("global_atomic_add_f32 %0, %1, off scope:SCOPE_DEV"
               :
               : "v"(p), "v"(v)
               : "memory");
}

// ---------------------------------------------------------------------------
// Kernel 3: agg[dst[e]] += h[src[e]]  (segment_sum over edges)
// One wave per edge: lane c handles float4 chunk c of the 128-float row
// (32 lanes * 4 floats = 128). h rows and agg both fit in the 192 MB L2, so
// global_atomic_add_f32 resolves at L2 speed. The edge index is wave-uniform;
// readfirstlane forces src/dst reads onto the scalar (SMEM) path.
// ---------------------------------------------------------------------------
__global__ __launch_bounds__(256) void edge_scatter_add(
    const float* __restrict__ h, const int* __restrict__ src,
    const int* __restrict__ dst, float* __restrict__ agg, int n_edges) {
  const long t = (long)blockIdx.x * blockDim.x + threadIdx.x;
  const int e = __builtin_amdgcn_readfirstlane((int)(t >> 5));  // wave-uniform edge id
  const int c = (int)(t & 31);                                   // float4 chunk 0..31
  if (e >= n_edges) return;
  const int s = src[e];   // SGPR index -> s_load_b32
  const int d = dst[e];
  const float4 v = *reinterpret_cast<const float4*>(h + (size_t)s * FEATS + c * 4);
  float* base = agg + (size_t)d * FEATS + c * 4;
  atomic_add_f32_dev(base + 0, v.x);
  atomic_add_f32_dev(base + 1, v.y);
  atomic_add_f32_dev(base + 2, v.z);
  atomic_add_f32_dev(base + 3, v.w);
}

extern "C" void kernel_launch(void* const* d_in, const int* in_sizes, int n_in,
                              void* d_out, int out_size, void* d_ws, size_t ws_size,
                              hipStream_t stream) {
  const float* h    = (const float*)d_in[0];   // [40000, 128]
  const float* W    = (const float*)d_in[1];   // [128, 128]
  const float* bias = (const float*)d_in[2];   // [128]
  const int*   src  = (const int*)d_in[3];     // [640000]
  const int*   dst  = (const int*)d_in[4];     // [640000]
  const int n_edges = in_sizes[3];

  float* out = (float*)d_out;                          // output 0: [40000,128]
  float* agg = out + (size_t)N_NODES * FEATS;          // output 1: [40000,128]

  // Zero agg (stream order guarantees completion before the scatter).
  const long n4 = (long)N_NODES * FEATS / 4;
  zero_f32<<<1024, 256, 0, stream>>>(agg, n4);

  // GEMM + bias via fp32 WMMA.
  gcn_gemm_wmma<<<N_NODES / TILE, 256, 0, stream>>>(h, W, bias, out);

  // Edge gather / scatter-add (one wave per edge).
  const long threads = (long)n_edges * 32;
  const int blocks = (int)((threads + 255) / 256);
  edge_scatter_add<<<blocks, 256, 0, stream>>>(h, src, dst, agg, n_edges);
}